// SubglacialDrainageSystem_30605936951815
// MI455X (gfx1250) — compile-verified
//
#include <hip/hip_runtime.h>
#include <hip/hip_bf16.h>

#define NN 4096      // number of nodes (matrix dimension)
#define NBK 128      // LU panel/block size
#define KSHEET 0.01f

typedef __attribute__((ext_vector_type(2))) float v2f;
typedef __attribute__((ext_vector_type(8))) float v8f;

// ---------------------------------------------------------------------------
// 1) Zero the dense matrix (L2-resident, 64MB)
// ---------------------------------------------------------------------------
__global__ void zero_mat(float4* Lv, long total4) {
    long i = (long)blockIdx.x * blockDim.x + threadIdx.x;
    long stride = (long)gridDim.x * blockDim.x;
    float4 z = make_float4(0.f, 0.f, 0.f, 0.f);
    for (; i < total4; i += stride) Lv[i] = z;
}

// ---------------------------------------------------------------------------
// 2) Assemble sparse 5-point operator into dense L, and forcing f into out
// ---------------------------------------------------------------------------
__global__ void assemble(const float* __restrict__ h, const float* __restrict__ phi,
                         const float* __restrict__ base, const float* __restrict__ over,
                         const float* __restrict__ melt, const float* __restrict__ len,
                         const int* __restrict__ adj, const int* __restrict__ inflow,
                         float* __restrict__ L, float* __restrict__ f) {
    int i = blockIdx.x * blockDim.x + threadIdx.x;
    if (i >= NN) return;
    bool isin = (inflow[i] == 1);
    float hi = h[i], pi = phi[i];
    float diag = 0.f;
    for (int s = 0; s < 4; ++s) {
        int j = adj[i * 4 + s];
        float ln = len[i * 4 + s];
        bool valid = (j >= 0) && (ln > 0.f);
        int jj = valid ? j : 0;
        float safe_len = valid ? ln : 1.f;
        float hf = 0.5f * (hi + h[jj]);
        float dphi = fabsf(pi - phi[jj]);
        float root = rsqrtf(fmaxf(dphi, 1e-12f) / safe_len);   // |grad phi|^-1/2
        float coeff = -KSHEET * powf(hf, 1.25f) * ln * root;
        coeff = (valid && !isin) ? coeff : 0.f;
        L[(long)i * NN + jj] += coeff;       // row i owned by this thread only
        diag -= coeff;
    }
    L[(long)i * NN + i] = isin ? 1.f : diag;
    f[i] = isin ? (base[i] - over[i]) : melt[i];
}

// ---------------------------------------------------------------------------
// 3) LU of the NBKxNBK diagonal block, in LDS (stored transposed so the
//    per-row updates S_T[j][i=tid] are bank-conflict-free across tid).
//    No pivoting: L is weakly row-diagonally-dominant / identity rows.
// ---------------------------------------------------------------------------
__global__ void lu_diag(float* __restrict__ L, int k0) {
    __shared__ float S[NBK * NBK];           // S[j*NBK + i] = A[i][j], 64KB
    int tid = threadIdx.x;                   // 128 threads
    for (int i = 0; i < NBK; ++i)
        S[tid * NBK + i] = L[(long)(k0 + i) * NN + k0 + tid];
    __syncthreads();
    for (int kk = 0; kk < NBK - 1; ++kk) {
        float pivot = S[kk * NBK + kk];
        float lik = 0.f;
        if (tid > kk) lik = S[kk * NBK + tid] / pivot;   // A[tid][kk]
        __syncthreads();
        if (tid > kk) {
            S[kk * NBK + tid] = lik;
            for (int j = kk + 1; j < NBK; ++j)
                S[j * NBK + tid] -= lik * S[j * NBK + kk];   // A[tid][j] -= lik*A[kk][j]
        }
        __syncthreads();
    }
    for (int i = 0; i < NBK; ++i)
        L[(long)(k0 + i) * NN + k0 + tid] = S[tid * NBK + i];
}

// ---------------------------------------------------------------------------
// 4) L21 = A21 * U11^{-1}.  One block = 128 trailing rows, one row per thread.
// ---------------------------------------------------------------------------
__global__ void trsm_l21(float* __restrict__ L, int k0) {
    __shared__ float XT[NBK * NBK];          // XT[j*NBK + r]
    __shared__ float Ucol[NBK];
    int tid = threadIdx.x;                   // 128 threads
    long r = (long)k0 + NBK + (long)blockIdx.x * NBK + tid;
    float* Arow = L + r * NN + k0;
    for (int j = 0; j < NBK; j += 4) {
        float4 v = *(const float4*)(Arow + j);
        XT[(j + 0) * NBK + tid] = v.x; XT[(j + 1) * NBK + tid] = v.y;
        XT[(j + 2) * NBK + tid] = v.z; XT[(j + 3) * NBK + tid] = v.w;
    }
    for (int j = 0; j < NBK; ++j) {
        if (tid <= j) Ucol[tid] = L[(long)(k0 + tid) * NN + k0 + j];
        __syncthreads();
        float xv = XT[j * NBK + tid];
        for (int t = 0; t < j; ++t)
            xv -= XT[t * NBK + tid] * Ucol[t];
        XT[j * NBK + tid] = xv / Ucol[j];
        __syncthreads();
    }
    for (int j = 0; j < NBK; j += 4) {
        float4 v = make_float4(XT[(j + 0) * NBK + tid], XT[(j + 1) * NBK + tid],
                               XT[(j + 2) * NBK + tid], XT[(j + 3) * NBK + tid]);
        *(float4*)(Arow + j) = v;
    }
}

// ---------------------------------------------------------------------------
// 5) U12 = L11^{-1} * A12 (unit-lower).  One block = 128 trailing columns.
// ---------------------------------------------------------------------------
__global__ void trsm_u12(float* __restrict__ L, int k0) {
    __shared__ float X[NBK * NBK];           // X[i*NBK + c]
    __shared__ float Lrow[NBK];
    int tid = threadIdx.x;                   // 128 threads
    long c0 = (long)k0 + NBK + (long)blockIdx.x * NBK;
    for (int i = 0; i < NBK; ++i)
        X[i * NBK + tid] = L[(long)(k0 + i) * NN + c0 + tid];   // coalesced
    for (int i = 1; i < NBK; ++i) {
        if (tid < i) Lrow[tid] = L[(long)(k0 + i) * NN + k0 + tid];  // coalesced
        __syncthreads();
        float xv = X[i * NBK + tid];
        for (int t = 0; t < i; ++t)
            xv -= Lrow[t] * X[t * NBK + tid];
        X[i * NBK + tid] = xv;
        __syncthreads();
    }
    for (int i = 0; i < NBK; ++i)
        L[(long)(k0 + i) * NN + c0 + tid] = X[i * NBK + tid];    // coalesced
}

// ---------------------------------------------------------------------------
// 6) Trailing update A22 -= L21 * U12 via V_WMMA_F32_16X16X4_F32,
//    2x2 register-blocked: each wave owns a 32x32 C block (4 fragments).
//    Each A fragment feeds 2 WMMAs, each B fragment feeds 2 WMMAs:
//    6 VMEM per 4 WMMAs (8 FLOP/byte from L2) instead of 3 VMEM per WMMA.
//    Fragment layouts per ISA 7.12.2 (32-bit A 16x4: lanes 0-15 K=0,1;
//    lanes 16-31 K=2,3 -> contiguous pair -> one 8B load).
// ---------------------------------------------------------------------------
__global__ void gemm_update(float* __restrict__ L, int k0, int m) {
    int lane = threadIdx.x & 31;
    int wave = threadIdx.x >> 5;
    int ntile = m >> 5;                                   // 32x32 tiles per side
    long tileIdx = (long)blockIdx.x * (blockDim.x >> 5) + wave;   // wave-uniform
    long total = (long)ntile * ntile;
    if (tileIdx >= total) return;    // uniform per wave -> EXEC all-1s for WMMA
    int tm = (int)(tileIdx / ntile);
    int tn = (int)(tileIdx % ntile);
    long t0 = (long)k0 + NBK;
    long r0 = t0 + (long)tm * 32;
    long c0 = t0 + (long)tn * 32;
    int lhi = lane >> 4;      // 0: lanes 0-15, 1: lanes 16-31
    int llo = lane & 15;

    // C fragments: VGPR r of cXY holds C[16*X + r + 8*lhi][16*Y + llo]
    v8f c00, c01, c10, c11;
    long crow = r0 + 8 * lhi;
    for (int r = 0; r < 8; ++r) {
        c00[r] = L[(crow + r) * NN + c0 + llo];
        c01[r] = L[(crow + r) * NN + c0 + 16 + llo];
        c10[r] = L[(crow + 16 + r) * NN + c0 + llo];
        c11[r] = L[(crow + 16 + r) * NN + c0 + 16 + llo];
    }

    const float* A0 = L + (r0 + llo) * (long)NN + k0 + 2 * lhi;        // rows 0-15
    const float* A1 = L + (r0 + 16 + llo) * (long)NN + k0 + 2 * lhi;   // rows 16-31
    const float* B0 = L + ((long)k0 + 2 * lhi) * NN + c0 + llo;        // cols 0-15
    const float* B1 = L + ((long)k0 + 2 * lhi) * NN + c0 + 16 + llo;   // cols 16-31

    for (int kk = 0; kk < NBK; kk += 4) {
        float2 av0 = *(const float2*)(A0 + kk);     // contiguous K pair, 8B aligned
        float2 av1 = *(const float2*)(A1 + kk);
        v2f a0; a0[0] = -av0.x; a0[1] = -av0.y;     // negate A => C - A*B
        v2f a1; a1[0] = -av1.x; a1[1] = -av1.y;
        v2f b0, b1;
        long bo = (long)kk * NN;
        b0[0] = B0[bo];      b0[1] = B0[bo + NN];
        b1[0] = B1[bo];      b1[1] = B1[bo + NN];
        c00 = __builtin_amdgcn_wmma_f32_16x16x4_f32(false, a0, false, b0, (short)0, c00, false, false);
        c01 = __builtin_amdgcn_wmma_f32_16x16x4_f32(false, a0, false, b1, (short)0, c01, false, false);
        c10 = __builtin_amdgcn_wmma_f32_16x16x4_f32(false, a1, false, b0, (short)0, c10, false, false);
        c11 = __builtin_amdgcn_wmma_f32_16x16x4_f32(false, a1, false, b1, (short)0, c11, false, false);
    }
    for (int r = 0; r < 8; ++r) {
        L[(crow + r) * NN + c0 + llo]           = c00[r];
        L[(crow + r) * NN + c0 + 16 + llo]      = c01[r];
        L[(crow + 16 + r) * NN + c0 + llo]      = c10[r];
        L[(crow + 16 + r) * NN + c0 + 16 + llo] = c11[r];
    }
}

// ---------------------------------------------------------------------------
// Wave-level sum (wave32 lane-permute tree), then tiny LDS combine
// ---------------------------------------------------------------------------
__device__ __forceinline__ float wave_sum(float s) {
    for (int off = 16; off > 0; off >>= 1)
        s += __shfl_xor(s, off, 32);
    return s;
}

// ---------------------------------------------------------------------------
// 7) Forward substitution L y = f (unit lower), RHS resident in LDS (16KB)
// ---------------------------------------------------------------------------
__global__ void fwd_solve(const float* __restrict__ L, float* __restrict__ y) {
    __shared__ float ys[NN];
    __shared__ float red[8];
    int tid = threadIdx.x;
    int lane = tid & 31, wave = tid >> 5;
    for (int i = tid; i < NN; i += 256) ys[i] = y[i];
    __syncthreads();
    for (int i = 1; i < NN; ++i) {
        const float* row = L + (long)i * NN;
        float s = 0.f;
        for (int j = tid; j < i; j += 256) s += row[j] * ys[j];
        s = wave_sum(s);
        if (lane == 0) red[wave] = s;
        __syncthreads();
        if (tid == 0) {
            float t = 0.f;
            for (int w = 0; w < 8; ++w) t += red[w];
            ys[i] -= t;
        }
        __syncthreads();
    }
    for (int i = tid; i < NN; i += 256) y[i] = ys[i];
}

// ---------------------------------------------------------------------------
// 8) Backward substitution U x = y
// ---------------------------------------------------------------------------
__global__ void bwd_solve(const float* __restrict__ L, float* __restrict__ x) {
    __shared__ float xs[NN];
    __shared__ float red[8];
    int tid = threadIdx.x;
    int lane = tid & 31, wave = tid >> 5;
    for (int i = tid; i < NN; i += 256) xs[i] = x[i];
    __syncthreads();
    for (int i = NN - 1; i >= 0; --i) {
        const float* row = L + (long)i * NN;
        float s = 0.f;
        for (int j = i + 1 + tid; j < NN; j += 256) s += row[j] * xs[j];
        s = wave_sum(s);
        if (lane == 0) red[wave] = s;
        __syncthreads();
        if (tid == 0) {
            float t = 0.f;
            for (int w = 0; w < 8; ++w) t += red[w];
            xs[i] = (xs[i] - t) / row[i];
        }
        __syncthreads();
    }
    for (int i = tid; i < NN; i += 256) x[i] = xs[i];
}

// ---------------------------------------------------------------------------
extern "C" void kernel_launch(void* const* d_in, const int* in_sizes, int n_in,
                              void* d_out, int out_size, void* d_ws, size_t ws_size,
                              hipStream_t stream) {
    const float* h    = (const float*)d_in[0];
    const float* phi  = (const float*)d_in[1];
    const float* base = (const float*)d_in[2];
    const float* over = (const float*)d_in[3];
    const float* melt = (const float*)d_in[4];
    const float* len  = (const float*)d_in[5];
    const int*   adj  = (const int*)d_in[6];
    const int*   infl = (const int*)d_in[7];
    float* L = (float*)d_ws;             // 4096*4096*4 = 64MB dense matrix
    float* out = (float*)d_out;          // forcing -> y -> x, in place

    long total4 = ((long)NN * NN) / 4;
    zero_mat<<<2048, 256, 0, stream>>>((float4*)L, total4);
    assemble<<<(NN + 255) / 256, 256, 0, stream>>>(h, phi, base, over, melt,
                                                   len, adj, infl, L, out);

    for (int kb = 0; kb < NN / NBK; ++kb) {
        int k0 = kb * NBK;
        lu_diag<<<1, NBK, 0, stream>>>(L, k0);
        int m = NN - k0 - NBK;
        if (m > 0) {
            trsm_l21<<<m / NBK, NBK, 0, stream>>>(L, k0);
            trsm_u12<<<m / NBK, NBK, 0, stream>>>(L, k0);
            long ntile = m / 32;                      // 32x32 per wave
            long tiles = ntile * ntile;
            int wpb = 8;                              // 256 threads = 8 waves
            long blocks = (tiles + wpb - 1) / wpb;
            gemm_update<<<dim3((unsigned)blocks), 256, 0, stream>>>(L, k0, m);
        }
    }

    fwd_solve<<<1, 256, 0, stream>>>(L, out);
    bwd_solve<<<1, 256, 0, stream>>>(L, out);
}